// GNN_82008105549934
// MI455X (gfx1250) — compile-verified
//
#include <hip/hip_runtime.h>

// ---------------------------------------------------------------------------
// GNN forward for MI455X (gfx1250).
//   - fp32 WMMA (V_WMMA_F32_16X16X4_F32) for all GEMMs (reference is fp32).
//   - Edge aggregation AFTER the W_rel GEMM (linearity) -> scatter width is
//     cout (<=128); [V,128] arrays stay resident in the 192MB L2.
//   - W staged in double-buffered LDS (zero-padded past K); B reads are
//     ds_load with immediate offsets -> no global-load stalls per WMMA.
// ---------------------------------------------------------------------------

#define VOXN 64

typedef float v2f __attribute__((ext_vector_type(2)));
typedef float v8f __attribute__((ext_vector_type(8)));

__device__ __forceinline__ void atomAddF(float* p, float v) {
  __hip_atomic_fetch_add(p, v, __ATOMIC_RELAXED, __HIP_MEMORY_SCOPE_AGENT);
}

__device__ __forceinline__ v8f wmma4(v2f a, v2f b, v8f c) {
  return __builtin_amdgcn_wmma_f32_16x16x4_f32(false, a, false, b, (short)0, c,
                                               false, false);
}

// ---------------------------------------------------------------------------
// Out[V,N] = A[V,K] @ W[K,N]  (+ Tacc*degr + bias in epilogue)
// Block = 8 waves; wave -> 32 rows (2 M-tiles) x NT*16 cols.
// A layout (ISA 16x4 f32): lanes 0-15: row=lane, v0=K0 v1=K1;
//                          lanes 16-31: v0=K2 v1=K3.  B mirrored.
// C/D: 8 VGPRs, lanes 0-15 -> M=r, lanes 16-31 -> M=r+8, N=lane&15.
// W chunks (KT x N) staged in LDS, zero-padded past K.  Requires K even.
// ---------------------------------------------------------------------------
template <int NT>
__global__ __launch_bounds__(256) void wmma_gemm(
    const float* __restrict__ A, const float* __restrict__ W,
    float* __restrict__ Out, const float* __restrict__ Tacc,
    const float* __restrict__ degr, const float* __restrict__ bias,
    int V, int K, int N) {
  constexpr int KT = 16;
  constexpr int LSTR = NT * 16 + 8;  // LDS row stride: 2*LSTR % 64 == 16
  __shared__ float sW[2][KT * LSTR];

  const int lane = threadIdx.x & 31;
  const int wave = blockIdx.x * (blockDim.x >> 5) + (threadIdx.x >> 5);
  int m0 = wave << 5;  // 32 rows per wave
  const bool dup = m0 >= V;          // keep barrier membership; skip stores
  if (dup) m0 = ((V - 1) >> 5) << 5;

  const int mr = lane & 15;
  const int kh = (lane >> 4) << 1;  // 0 | 2

  // Clamped rows/cols: clamped duplicates are never stored.
  int r0 = m0 + mr;      if (r0 > V - 1) r0 = V - 1;
  int r1 = m0 + 16 + mr; if (r1 > V - 1) r1 = V - 1;
  const float* a0p = A + (size_t)r0 * K + kh;
  const float* a1p = A + (size_t)r1 * K + kh;
  int lofs[NT];
#pragma unroll
  for (int t = 0; t < NT; ++t) {
    const int n = (t << 4) + mr;
    lofs[t] = kh * LSTR + (n < N ? n : N - 1);
  }

  v8f acc0[NT], acc1[NT];
#pragma unroll
  for (int t = 0; t < NT; ++t) {
    acc0[t] = v8f{0.f, 0.f, 0.f, 0.f, 0.f, 0.f, 0.f, 0.f};
    acc1[t] = v8f{0.f, 0.f, 0.f, 0.f, 0.f, 0.f, 0.f, 0.f};
  }

  // Cooperative W chunk staging (zero-padded past K).
  auto stage = [&](int k0, int bufi) {
    float* dst = &sW[bufi][0];
    if ((256 % N) == 0) {
      const int rstep = 256 / N;
      int r = (int)threadIdx.x / N;
      const int n = (int)threadIdx.x - r * N;
      for (; r < KT; r += rstep)
        dst[r * LSTR + n] = (k0 + r < K) ? W[(size_t)(k0 + r) * N + n] : 0.f;
    } else {
      for (int idx = threadIdx.x; idx < KT * N; idx += 256) {
        const int r = idx / N, n = idx - r * N;
        dst[r * LSTR + n] = (k0 + r < K) ? W[(size_t)(k0 + r) * N + n] : 0.f;
      }
    }
  };

  const int nch = (K + KT - 1) / KT;
  int buf = 0;
  for (int c = 0; c < nch; ++c, buf ^= 1) {
    const int k0 = c * KT;
    stage(k0, buf);
    __syncthreads();  // one barrier per chunk; double buffer covers WAR
    const float* sb = &sW[buf][0];
    const int krem = K - k0;
    if (krem >= KT) {
      __builtin_prefetch(a0p + k0 + KT, 0, 3);
      __builtin_prefetch(a1p + k0 + KT, 0, 3);
#pragma unroll
      for (int kl = 0; kl < KT; kl += 4) {
        const v2f a0 = *(const v2f*)(a0p + k0 + kl);
        const v2f a1 = *(const v2f*)(a1p + k0 + kl);
#pragma unroll
        for (int t = 0; t < NT; ++t) {
          v2f b;
          b.x = sb[kl * LSTR + lofs[t]];
          b.y = sb[kl * LSTR + lofs[t] + LSTR];
          acc0[t] = wmma4(a0, b, acc0[t]);
          acc1[t] = wmma4(a1, b, acc1[t]);
        }
      }
    } else {  // final partial chunk: mask A (B is zero-padded in LDS)
      for (int kl = 0; kl < krem; kl += 4) {
        const int ka = k0 + kl + kh;
        const bool k0ok = ka < K, k1ok = ka + 1 < K;
        v2f a0, a1;
        a0.x = k0ok ? a0p[ka] : 0.f;
        a0.y = k1ok ? a0p[ka + 1] : 0.f;
        a1.x = k0ok ? a1p[ka] : 0.f;
        a1.y = k1ok ? a1p[ka + 1] : 0.f;
#pragma unroll
        for (int t = 0; t < NT; ++t) {
          v2f b;
          b.x = sb[kl * LSTR + lofs[t]];
          b.y = sb[kl * LSTR + lofs[t] + LSTR];
          acc0[t] = wmma4(a0, b, acc0[t]);
          acc1[t] = wmma4(a1, b, acc1[t]);
        }
      }
    }
  }

  if (dup) return;
  const int rowoff = (lane >> 4) << 3;  // 0 | 8
#pragma unroll
  for (int t = 0; t < NT; ++t) {
    const int col = (t << 4) + mr;
    if (col >= N) continue;
#pragma unroll
    for (int r = 0; r < 8; ++r) {
      int row = m0 + r + rowoff;
      if (row < V) {
        float v = acc0[t][r];
        if (Tacc) v += Tacc[(size_t)row * N + col] * degr[row];
        if (bias) v += bias[col];
        Out[(size_t)row * N + col] = v;
      }
      row += 16;
      if (row < V) {
        float v = acc1[t][r];
        if (Tacc) v += Tacc[(size_t)row * N + col] * degr[row];
        if (bias) v += bias[col];
        Out[(size_t)row * N + col] = v;
      }
    }
  }
}

// ---------------------------------------------------------------------------
__global__ void deg_kernel(const int* __restrict__ edges, float* __restrict__ deg,
                           int E) {
  for (long long e = blockIdx.x * (long long)blockDim.x + threadIdx.x; e < E;
       e += (long long)gridDim.x * blockDim.x)
    atomAddF(&deg[edges[2 * e + 1]], 1.0f);
}

__global__ void deg_recip_kernel(float* __restrict__ deg, int V) {
  for (int i = blockIdx.x * blockDim.x + threadIdx.x; i < V;
       i += gridDim.x * blockDim.x)
    deg[i] = 1.f / fmaxf(deg[i], 1.f);
}

// Vectorized scatter: chunk = 4 channels, cshift = log2(C/4).
__global__ void scatter4_kernel(const float* __restrict__ P,
                                const int* __restrict__ edges,
                                float* __restrict__ T, int E, int cshift,
                                int C) {
  const long long total = (long long)E << cshift;
  const int cmask = (1 << cshift) - 1;
  for (long long i = blockIdx.x * (long long)blockDim.x + threadIdx.x; i < total;
       i += (long long)gridDim.x * blockDim.x) {
    const int e = (int)(i >> cshift);
    const int c = ((int)i & cmask) << 2;
    const int s = edges[2 * e], d = edges[2 * e + 1];
    const float4 v = *(const float4*)(P + (size_t)s * C + c);
    float* tp = T + (size_t)d * C + c;
    atomAddF(tp + 0, v.x);
    atomAddF(tp + 1, v.y);
    atomAddF(tp + 2, v.z);
    atomAddF(tp + 3, v.w);
  }
}

__global__ void scatter_kernel(const float* __restrict__ P,
                               const int* __restrict__ edges,
                               float* __restrict__ T, int E, int C) {
  const long long total = (long long)E * C;
  for (long long i = blockIdx.x * (long long)blockDim.x + threadIdx.x; i < total;
       i += (long long)gridDim.x * blockDim.x) {
    const int e = (int)(i / C);
    const int c = (int)(i - (long long)e * C);
    const int s = edges[2 * e], d = edges[2 * e + 1];
    atomAddF(&T[(size_t)d * C + c], P[(size_t)s * C + c]);
  }
}

// Row copy, cs power of two, float2 (ldx/col0/c all even at call sites).
__global__ void copy_cols2_kernel(const float* __restrict__ S, int cshift,
                                  float* __restrict__ X, int ldx, int col0,
                                  long long total2) {
  const int cmask = (1 << cshift) - 1;
  for (long long i = blockIdx.x * (long long)blockDim.x + threadIdx.x;
       i < total2; i += (long long)gridDim.x * blockDim.x) {
    const int v = (int)(i >> cshift);
    const int c = ((int)i & cmask) << 1;
    *(v2f*)(X + (size_t)v * ldx + col0 + c) =
        *(const v2f*)(S + ((size_t)v << (cshift + 1)) + c);
  }
}

__global__ void copy_cols_kernel(const float* __restrict__ S, int cs,
                                 float* __restrict__ X, int ldx, int col0,
                                 int V) {
  const long long total = (long long)V * cs;
  for (long long i = blockIdx.x * (long long)blockDim.x + threadIdx.x; i < total;
       i += (long long)gridDim.x * blockDim.x) {
    const int v = (int)(i / cs);
    const int c = (int)(i - (long long)v * cs);
    X[(size_t)v * ldx + col0 + c] = S[i];
  }
}

__global__ void add_relu_kernel(const float* __restrict__ H,
                                const float* __restrict__ R,
                                float* __restrict__ F, long long total) {
  for (long long i = blockIdx.x * (long long)blockDim.x + threadIdx.x; i < total;
       i += (long long)gridDim.x * blockDim.x)
    F[i] = fmaxf(H[i] + R[i], 0.f);
}

// Trilinear sampling at clamp(verts + sgn*normals, 0, 63) (reference's grid
// normalization is the identity since sizes == vshape).  cshift = log2(C).
__global__ void trilerp_kernel(const float* __restrict__ vf, int cshift,
                               const float* __restrict__ verts,
                               const float* __restrict__ nrm, float sgn,
                               float* __restrict__ X, int ldx, int col0, int V) {
  const long long total = (long long)V << cshift;
  const int cmask = (1 << cshift) - 1;
  for (long long i = blockIdx.x * (long long)blockDim.x + threadIdx.x; i < total;
       i += (long long)gridDim.x * blockDim.x) {
    const int v = (int)(i >> cshift);
    const int c = (int)i & cmask;
    float pd = verts[3 * v + 0] + sgn * nrm[3 * v + 0];
    float ph = verts[3 * v + 1] + sgn * nrm[3 * v + 1];
    float pw = verts[3 * v + 2] + sgn * nrm[3 * v + 2];
    pd = fminf(fmaxf(pd, 0.f), 63.f);
    ph = fminf(fmaxf(ph, 0.f), 63.f);
    pw = fminf(fmaxf(pw, 0.f), 63.f);
    const int d0 = (int)pd, h0 = (int)ph, w0 = (int)pw;
    const float fd = pd - d0, fh = ph - h0, fw = pw - w0;
    const int d1 = d0 + 1 > 63 ? 63 : d0 + 1;
    const int h1 = h0 + 1 > 63 ? 63 : h0 + 1;
    const int w1 = w0 + 1 > 63 ? 63 : w0 + 1;
    const float* f = vf + ((size_t)c << 18);  // c * 64^3
#define AT(d, h, w) f[(((d)*VOXN) + (h)) * VOXN + (w)]
    const float c000 = AT(d0, h0, w0), c001 = AT(d0, h0, w1);
    const float c010 = AT(d0, h1, w0), c011 = AT(d0, h1, w1);
    const float c100 = AT(d1, h0, w0), c101 = AT(d1, h0, w1);
    const float c110 = AT(d1, h1, w0), c111 = AT(d1, h1, w1);
#undef AT
    const float gd = 1.f - fd, gh = 1.f - fh, gw = 1.f - fw;
    X[(size_t)v * ldx + col0 + c] =
        c000 * gd * gh * gw + c001 * gd * gh * fw + c010 * gd * fh * gw +
        c011 * gd * fh * fw + c100 * fd * gh * gw + c101 * fd * gh * fw +
        c110 * fd * fh * gw + c111 * fd * fh * fw;
  }
}

// Fused LayerNorm(128) + ReLU, one wave32 per row.
__global__ __launch_bounds__(256) void lnorm_relu128(
    float* __restrict__ X, const float* __restrict__ g,
    const float* __restrict__ b, int V) {
  const int lane = threadIdx.x & 31;
  const int row = blockIdx.x * (blockDim.x >> 5) + (threadIdx.x >> 5);
  if (row >= V) return;
  float* x = X + (size_t)row * 128;
  float v0 = x[lane], v1 = x[lane + 32], v2 = x[lane + 64], v3 = x[lane + 96];
  float s = v0 + v1 + v2 + v3;
#pragma unroll
  for (int off = 16; off; off >>= 1) s += __shfl_xor(s, off, 32);
  const float mu = s * (1.f / 128.f);
  const float d0 = v0 - mu, d1 = v1 - mu, d2 = v2 - mu, d3 = v3 - mu;
  float ss = d0 * d0 + d1 * d1 + d2 * d2 + d3 * d3;
#pragma unroll
  for (int off = 16; off; off >>= 1) ss += __shfl_xor(ss, off, 32);
  const float inv = rsqrtf(ss * (1.f / 128.f) + 1e-5f);
  x[lane]      = fmaxf(d0 * inv * g[lane]      + b[lane],      0.f);
  x[lane + 32] = fmaxf(d1 * inv * g[lane + 32] + b[lane + 32], 0.f);
  x[lane + 64] = fmaxf(d2 * inv * g[lane + 64] + b[lane + 64], 0.f);
  x[lane + 96] = fmaxf(d3 * inv * g[lane + 96] + b[lane + 96], 0.f);
}

// ---------------------------------------------------------------------------
// Host side
// ---------------------------------------------------------------------------
static inline int nblk(long long total, int tpb) {
  long long b = (total + tpb - 1) / tpb;
  if (b > 262144) b = 262144;
  if (b < 1) b = 1;
  return (int)b;
}

static void launch_gemm(int NT, const float* A, const float* W, float* Out,
                        const float* T, const float* degr, const float* bias,
                        int V, int K, int N, hipStream_t s) {
  const int waves = (V + 31) / 32;
  const int blocks = (waves + 7) / 8;  // 8 waves / 256-thread block
  if (NT == 8)
    wmma_gemm<8><<<blocks, 256, 0, s>>>(A, W, Out, T, degr, bias, V, K, N);
  else if (NT == 4)
    wmma_gemm<4><<<blocks, 256, 0, s>>>(A, W, Out, T, degr, bias, V, K, N);
  else
    wmma_gemm<1><<<blocks, 256, 0, s>>>(A, W, Out, T, degr, bias, V, K, N);
}

struct GCBp { const float *Wrel, *Wroot, *lnb, *lng; };
struct RBp  { GCBp l[2]; const float *linW, *linb; };

extern "C" void kernel_launch(void* const* d_in, const int* in_sizes, int n_in,
                              void* d_out, int out_size, void* d_ws,
                              size_t ws_size, hipStream_t stream) {
  const int V = in_sizes[0] / 3;
  const int E = in_sizes[1] / 2;

  const float* vertices = (const float*)d_in[0];
  const int*   edges    = (const int*)d_in[1];
  const float* normals  = (const float*)d_in[2];
  const float* vfs[3]   = {(const float*)d_in[3], (const float*)d_in[4],
                           (const float*)d_in[5]};
  const int    vfShift[3] = {4, 5, 6};  // log2(C): 16, 32, 64
  const float* vpot     = (const float*)d_in[6];

  // ---- params: jax tree-flatten leaf order (sorted dict keys) ----
  int pi = 7;
  auto nx = [&]() { return (const float*)d_in[pi++]; };
  RBp rb[2][2];
  for (int bi = 0; bi < 2; ++bi)
    for (int ri = 0; ri < 2; ++ri) {
      RBp& r = rb[bi][ri];
      for (int li = 0; li < 2; ++li) {  // gcb: gc{W_rel,W_root}, ln_b, ln_g
        r.l[li].Wrel = nx(); r.l[li].Wroot = nx();
        r.l[li].lnb  = nx(); r.l[li].lng   = nx();
      }
      if (ri == 0) { r.linW = nx(); r.linb = nx(); }
      else         { r.linW = nullptr; r.linb = nullptr; }
    }
  const float* fWrel  = nx(); const float* fWroot = nx(); const float* fB = nx();
  const float* lWrel  = nx(); const float* lWroot = nx();

  // ---- workspace carve ----
  char* w = (char*)d_ws;
  auto alloc = [&](size_t nf) { float* p = (float*)w; w += nf * sizeof(float); return p; };
  float* deg = alloc(V);          // becomes 1/max(deg,1)
  float* inp = alloc((size_t)V * 6);
  float* X   = alloc((size_t)V * 422);
  float* P   = alloc((size_t)V * 128);
  float* T   = alloc((size_t)V * 128);
  float* H1  = alloc((size_t)V * 128);
  float* R   = alloc((size_t)V * 128);
  float* F   = alloc((size_t)V * 128);

  // gconv: Out = segmean((A@Wrel)[src]->dst) + A@Wroot (+bias)
  auto gconv = [&](const float* A, int K, int N, int NT, const float* Wrel,
                   const float* Wroot, const float* bias, float* Out) {
    launch_gemm(NT, A, Wrel, P, nullptr, nullptr, nullptr, V, K, N, stream);
    hipMemsetAsync(T, 0, (size_t)V * N * sizeof(float), stream);
    if ((N & 3) == 0) {
      int cshift = 0;
      while ((4 << cshift) < N) ++cshift;  // log2(N/4)
      scatter4_kernel<<<nblk((long long)E << cshift, 256), 256, 0, stream>>>(
          P, edges, T, E, cshift, N);
    } else {
      scatter_kernel<<<nblk((long long)E * N, 256), 256, 0, stream>>>(
          P, edges, T, E, N);
    }
    launch_gemm(NT, A, Wroot, Out, T, deg, bias, V, K, N, stream);
  };

  // ---- degrees -> reciprocal ----
  hipMemsetAsync(deg, 0, (size_t)V * sizeof(float), stream);
  deg_kernel<<<nblk(E, 256), 256, 0, stream>>>(edges, deg, E);
  deg_recip_kernel<<<nblk(V, 256), 256, 0, stream>>>(deg, V);

  // ---- inp = [vertices | vector_potential] ----
  copy_cols_kernel<<<nblk((long long)V * 3, 256), 256, 0, stream>>>(vertices, 3, inp, 6, 0, V);
  copy_cols_kernel<<<nblk((long long)V * 3, 256), 256, 0, stream>>>(vpot, 3, inp, 6, 3, V);

  // ---- first gconv: 6 -> 64, into F ----
  gconv(inp, 6, 64, /*NT=*/4, fWrel, fWroot, fB, F);
  int cf = 64, cfshift = 6;

  const int aggidx[2][2] = {{0, 1}, {1, 2}};
  const int cins[2] = {214, 422};
  const float sgns[3] = {0.f, 1.f, -1.f};

  for (int bi = 0; bi < 2; ++bi) {
    const int cin = cins[bi];
    // build X = [feats | agg(p), agg(p+n), agg(p-n) | inp]   (ldx even)
    copy_cols2_kernel<<<nblk((long long)V * (cf / 2), 256), 256, 0, stream>>>(
        F, cfshift - 1, X, cin, 0, (long long)V * (cf / 2));
    int col = cf;
    for (int s = 0; s < 3; ++s)
      for (int j = 0; j < 2; ++j) {
        const int fi = aggidx[bi][j];
        trilerp_kernel<<<nblk((long long)V << vfShift[fi], 256), 256, 0, stream>>>(
            vfs[fi], vfShift[fi], vertices, normals, sgns[s], X, cin, col, V);
        col += 1 << vfShift[fi];
      }
    copy_cols_kernel<<<nblk((long long)V * 6, 256), 256, 0, stream>>>(inp, 6, X, cin, col, V);

    for (int ri = 0; ri < 2; ++ri) {
      const RBp& r = rb[bi][ri];
      const float* xin = (ri == 0) ? X : F;
      const int K0 = (ri == 0) ? cin : 128;
      if (r.linW)  // residual projection
        launch_gemm(8, xin, r.linW, R, nullptr, nullptr, r.linb, V, K0, 128, stream);
      // layer 0 -> H1 (act)
      gconv(xin, K0, 128, 8, r.l[0].Wrel, r.l[0].Wroot, nullptr, H1);
      lnorm_relu128<<<(V + 7) / 8, 256, 0, stream>>>(H1, r.l[0].lng, r.l[0].lnb, V);
      // layer 1 -> P (act); P intermediate is dead after scatter, safe to reuse
      gconv(H1, 128, 128, 8, r.l[1].Wrel, r.l[1].Wroot, nullptr, P);
      lnorm_relu128<<<(V + 7) / 8, 256, 0, stream>>>(P, r.l[1].lng, r.l[1].lnb, V);
      // F = relu(h + residual)
      add_relu_kernel<<<nblk((long long)V * 128, 256), 256, 0, stream>>>(
          P, r.linW ? R : F, F, (long long)V * 128);
    }
    cf = 128; cfshift = 7;
  }

  // ---- last gconv: 128 -> 3, straight into d_out ----
  gconv(F, 128, 3, /*NT=*/1, lWrel, lWroot, nullptr, (float*)d_out);
}